// Model_82197084110866
// MI455X (gfx1250) — compile-verified
//
#include <hip/hip_runtime.h>
#include <math.h>
#include <stdint.h>

// Problem constants (match reference)
constexpr int B = 4;
constexpr int N = 4096;
constexpr int C = 512;
constexpr int E = 65536;
constexpr int M = 1024;
constexpr int H = 8;
constexpr int D = 64;
constexpr float NEG_SLOPE = 0.2f;

typedef __attribute__((ext_vector_type(2))) float v2f;
typedef __attribute__((ext_vector_type(8))) float v8f;
typedef int v4i __attribute__((vector_size(16)));   // type expected by the
                                                    // async-to-LDS builtin

// ---------------------------------------------------------------------------
// gfx1250 async global->LDS copy (ASYNCcnt-tracked), guarded so compilation
// cannot break if the builtin is absent in this toolchain.
// ---------------------------------------------------------------------------
#if __has_builtin(__builtin_amdgcn_global_load_async_to_lds_b128)
#define ASYNC_LDS 1
#else
#define ASYNC_LDS 0
#endif

__device__ __forceinline__ void stage_b128(const float* gsrc, float* ldst) {
#if ASYNC_LDS
    __builtin_amdgcn_global_load_async_to_lds_b128((v4i*)gsrc, (v4i*)ldst, 0, 0);
#else
    float4 t = *(const float4*)gsrc;
    *(float4*)ldst = t;
#endif
}

__device__ __forceinline__ void wait_stage() {
#if ASYNC_LDS
#if __has_builtin(__builtin_amdgcn_s_wait_asynccnt)
    __builtin_amdgcn_s_wait_asynccnt(0);
#else
    asm volatile("s_wait_asynccnt 0" ::: "memory");
#endif
#endif
    __syncthreads();
}

// ---------------------------------------------------------------------------
// Float atomic max via signed/unsigned integer atomics (standard bit trick).
// ---------------------------------------------------------------------------
__device__ __forceinline__ void atomicMaxF(float* addr, float val) {
    if (val >= 0.0f) {
        atomicMax((int*)addr, __float_as_int(val));
    } else {
        atomicMin((unsigned int*)addr, __float_as_uint(val));
    }
}

// ---------------------------------------------------------------------------
// Fill kernel (grid-stride)
// ---------------------------------------------------------------------------
__global__ void fill_f32(float* __restrict__ p, float v, size_t n) {
    size_t i = (size_t)blockIdx.x * blockDim.x + threadIdx.x;
    size_t stride = (size_t)gridDim.x * blockDim.x;
    for (; i < n; i += stride) p[i] = v;
}

// ---------------------------------------------------------------------------
// GEMM: XW[b] = X[b] (N x C) * W (C x C), fp32 WMMA 16x16x4.
// Block = 256 threads = 8 waves -> 128x64 macro-tile.
// Wave w computes rows [w*16, w*16+16) x 64 cols = 4 accumulators; the A
// fragment is reused across the 4 column tiles. W k-slab (32x64) is staged
// once per block into LDS (async-to-LDS when available) and shared by all
// 8 waves.
// wave32 fp32 WMMA layout:
//   A (16x4): lane group g=lane/16, row m=lane%16, a[v] = X[m][k+2g+v]
//   B (4x16): col n=lane%16,                       b[v] = W[k+2g+v][n]
//   C/D:      acc[v] -> row = base + v + 8g, col = base + lane%16
// ---------------------------------------------------------------------------
constexpr int KS = 32;       // k-slab depth staged in LDS
constexpr int LDSP = 72;     // padded LDS row pitch (floats): 16B-aligned,
                             // de-conflicts banks between lane halves (k vs k+2)

__global__ void gemm_wmma_f32(const float* __restrict__ X,
                              const float* __restrict__ W,
                              float* __restrict__ XW) {
    __shared__ float sW[KS * LDSP];

    constexpr int TMB = N / 128;              // 32 row macro-tiles per batch
    constexpr int TNB = C / 64;               // 8 col macro-tiles per batch
    const int bid = blockIdx.x;
    const int b   = bid / (TMB * TNB);
    const int r   = bid % (TMB * TNB);
    const int tmb = r / TNB;
    const int tnb = r % TNB;

    const int wave = threadIdx.x >> 5;
    const int lane = threadIdx.x & 31;
    const int g    = lane >> 4;               // 0 or 1
    const int hl   = lane & 15;

    const float* Xb  = X  + (size_t)b * N * C;
    float*       XWb = XW + (size_t)b * N * C;

    const int row_base = tmb * 128 + wave * 16;
    const int col_base = tnb * 64;
    const float* xrow = Xb + (size_t)(row_base + hl) * C;

    v8f acc[4] = {};

    for (int k0 = 0; k0 < C; k0 += KS) {
        // ---- stage W[k0:k0+32, col_base:col_base+64] into LDS ------------
        // 512 float4 chunks; 256 threads x 2 chunks.
        {
            int c0 = threadIdx.x;             // chunk ids c0 and c0+256
            #pragma unroll
            for (int cc = 0; cc < 2; ++cc) {
                int chunk = c0 + cc * 256;
                int kr   = chunk >> 4;        // 0..31
                int colc = (chunk & 15) * 4;  // 0..60
                stage_b128(W + (size_t)(k0 + kr) * C + col_base + colc,
                           &sW[kr * LDSP + colc]);
            }
        }
        wait_stage();

        // ---- compute on the slab ----------------------------------------
        #pragma unroll
        for (int kk = 0; kk < KS; kk += 4) {
            const int ka = kk + 2 * g;
            v2f a;
            a.x = xrow[k0 + ka];
            a.y = xrow[k0 + ka + 1];
            #pragma unroll
            for (int c = 0; c < 4; ++c) {
                v2f bb;
                bb.x = sW[ka * LDSP + c * 16 + hl];
                bb.y = sW[(ka + 1) * LDSP + c * 16 + hl];
                acc[c] = __builtin_amdgcn_wmma_f32_16x16x4_f32(
                    false, a, false, bb, (short)0, acc[c], false, false);
            }
        }
        __syncthreads();   // protect LDS before next stage overwrites it
    }

    #pragma unroll
    for (int c = 0; c < 4; ++c) {
        #pragma unroll
        for (int v = 0; v < 8; ++v) {
            XWb[(size_t)(row_base + v + 8 * g) * C + col_base + c * 16 + hl] =
                acc[c][v];
        }
    }
}

// ---------------------------------------------------------------------------
// Node / hyperedge degrees (as float, 1.0 per incidence)
// ---------------------------------------------------------------------------
__global__ void degrees_kernel(const int* __restrict__ hidx,
                               float* __restrict__ Dnode,
                               float* __restrict__ Bdeg) {
    size_t i = (size_t)blockIdx.x * blockDim.x + threadIdx.x; // over B*E
    int e = (int)(i % E);
    int b = (int)(i / E);
    const int* hb = hidx + (size_t)b * 2 * E;
    int s = hb[e];
    int t = hb[E + e];
    atomicAdd(&Dnode[(size_t)b * N + s], 1.0f);
    atomicAdd(&Bdeg[(size_t)b * M + t], 1.0f);
}

// ---------------------------------------------------------------------------
// Attention logits: one wave per (b, e). b128 gathers of x_i / x_j; each lane
// handles 4 consecutive channels (always within one head since 4 | 64).
// For pass t: lanes 0-15 -> head 2t, lanes 16-31 -> head 2t+1; width-16
// shuffle reduction finishes two heads per pass. Leaky-relu, then atomic
// segment-max keyed by src.
// ---------------------------------------------------------------------------
__global__ void alpha_logits(const float* __restrict__ XW,
                             const int* __restrict__ hidx,
                             const float* __restrict__ att,
                             float* __restrict__ alpha,
                             float* __restrict__ seg_max) {
    __shared__ float s_att[H * 2 * D]; // 1024 floats
    for (int i = threadIdx.x; i < H * 2 * D; i += blockDim.x) s_att[i] = att[i];
    __syncthreads();

    const int wid  = (int)(((size_t)blockIdx.x * blockDim.x + threadIdx.x) >> 5);
    const int lane = threadIdx.x & 31;
    const int b = wid / E;
    const int e = wid % E;
    const int* hb = hidx + (size_t)b * 2 * E;
    const int s = hb[e];
    const int t = hb[E + e];

    const float* xi = XW + ((size_t)b * N + s) * C;
    const float* xj = XW + ((size_t)b * N + t) * C;
    __builtin_prefetch(xi + lane * 4, 0, 0);
    __builtin_prefetch(xj + lane * 4, 0, 0);

    float* al = alpha + ((size_t)b * E + e) * H;
    float* sm = seg_max + ((size_t)b * N + s) * H;

    #pragma unroll
    for (int t4 = 0; t4 < 4; ++t4) {
        const int i  = lane * 4 + 128 * t4;
        const int h  = (lane >> 4) + 2 * t4;
        const int d  = (lane & 15) * 4;
        const float4 a4 = *(const float4*)(xi + i);
        const float4 b4 = *(const float4*)(xj + i);
        const float4 ai = *(const float4*)(s_att + h * 2 * D + d);
        const float4 aj = *(const float4*)(s_att + h * 2 * D + D + d);
        float p = a4.x * ai.x + a4.y * ai.y + a4.z * ai.z + a4.w * ai.w
                + b4.x * aj.x + b4.y * aj.y + b4.z * aj.z + b4.w * aj.w;
        #pragma unroll
        for (int off = 8; off > 0; off >>= 1) p += __shfl_down(p, off, 16);
        if ((lane & 15) == 0) {
            p = p > 0.0f ? p : NEG_SLOPE * p; // leaky relu
            al[h] = p;
            atomicMaxF(&sm[h], p);
        }
    }
}

// ---------------------------------------------------------------------------
// e = exp(logit - segmax); accumulate segment sums. Thread per (b,e,h).
// ---------------------------------------------------------------------------
__global__ void exp_and_sum(const int* __restrict__ hidx,
                            float* __restrict__ alpha,
                            const float* __restrict__ seg_max,
                            float* __restrict__ seg_sum) {
    size_t i = (size_t)blockIdx.x * blockDim.x + threadIdx.x; // over B*E*H
    int h = (int)(i % H);
    size_t be = i / H;
    int e = (int)(be % E);
    int b = (int)(be / E);
    int s = hidx[(size_t)b * 2 * E + e];
    float m = seg_max[((size_t)b * N + s) * H + h];
    if (!isfinite(m)) m = 0.0f;
    float ev = __expf(alpha[i] - m);
    alpha[i] = ev;
    atomicAdd(&seg_sum[((size_t)b * N + s) * H + h], ev);
}

// ---------------------------------------------------------------------------
// alpha /= max(segsum, 1e-16). Thread per (b,e,h).
// ---------------------------------------------------------------------------
__global__ void normalize_alpha(const int* __restrict__ hidx,
                                float* __restrict__ alpha,
                                const float* __restrict__ seg_sum) {
    size_t i = (size_t)blockIdx.x * blockDim.x + threadIdx.x;
    int h = (int)(i % H);
    size_t be = i / H;
    int e = (int)(be % E);
    int b = (int)(be / E);
    int s = hidx[(size_t)b * 2 * E + e];
    float ssum = seg_sum[((size_t)b * N + s) * H + h];
    alpha[i] = alpha[i] / fmaxf(ssum, 1e-16f);
}

// ---------------------------------------------------------------------------
// msg1: e_feat[edg] += Bnorm[edg] * x_i * alpha. One wave per (b,e); b128
// reads, per-float atomic adds.
// ---------------------------------------------------------------------------
__global__ void scatter_msg1(const float* __restrict__ XW,
                             const int* __restrict__ hidx,
                             const float* __restrict__ alpha,
                             const float* __restrict__ Bdeg,
                             float* __restrict__ e_feat) {
    const int wid  = (int)(((size_t)blockIdx.x * blockDim.x + threadIdx.x) >> 5);
    const int lane = threadIdx.x & 31;
    const int b = wid / E;
    const int e = wid % E;
    const int* hb = hidx + (size_t)b * 2 * E;
    const int s = hb[e];
    const int t = hb[E + e];

    float bd = Bdeg[(size_t)b * M + t];
    float bnorm = bd > 0.0f ? 1.0f / bd : 0.0f;

    const float* xi = XW + ((size_t)b * N + s) * C;
    const float* al = alpha + ((size_t)b * E + e) * H;
    float* ef = e_feat + ((size_t)b * M + t) * C;

    #pragma unroll
    for (int t4 = 0; t4 < 4; ++t4) {
        const int i = lane * 4 + 128 * t4;       // 4 channels, one head
        const float4 xv = *(const float4*)(xi + i);
        const float a = bnorm * al[i >> 6];
        atomicAdd(&ef[i + 0], xv.x * a);
        atomicAdd(&ef[i + 1], xv.y * a);
        atomicAdd(&ef[i + 2], xv.z * a);
        atomicAdd(&ef[i + 3], xv.w * a);
    }
}

// ---------------------------------------------------------------------------
// msg2: out[src] += Dnode[src] * e_feat[edg] * alpha. One wave per (b,e).
// ---------------------------------------------------------------------------
__global__ void scatter_msg2(const float* __restrict__ e_feat,
                             const int* __restrict__ hidx,
                             const float* __restrict__ alpha,
                             const float* __restrict__ Dnode,
                             float* __restrict__ out) {
    const int wid  = (int)(((size_t)blockIdx.x * blockDim.x + threadIdx.x) >> 5);
    const int lane = threadIdx.x & 31;
    const int b = wid / E;
    const int e = wid % E;
    const int* hb = hidx + (size_t)b * 2 * E;
    const int s = hb[e];
    const int t = hb[E + e];

    float dn = Dnode[(size_t)b * N + s];
    const float* ef = e_feat + ((size_t)b * M + t) * C;
    const float* al = alpha + ((size_t)b * E + e) * H;
    float* ob = out + ((size_t)b * N + s) * C;

    #pragma unroll
    for (int t4 = 0; t4 < 4; ++t4) {
        const int i = lane * 4 + 128 * t4;
        const float4 ev = *(const float4*)(ef + i);
        const float a = dn * al[i >> 6];
        atomicAdd(&ob[i + 0], ev.x * a);
        atomicAdd(&ob[i + 1], ev.y * a);
        atomicAdd(&ob[i + 2], ev.z * a);
        atomicAdd(&ob[i + 3], ev.w * a);
    }
}

// ---------------------------------------------------------------------------
// Launch
// ---------------------------------------------------------------------------
extern "C" void kernel_launch(void* const* d_in, const int* in_sizes, int n_in,
                              void* d_out, int out_size, void* d_ws, size_t ws_size,
                              hipStream_t stream) {
    const float* x    = (const float*)d_in[0]; // (B,N,C)
    const int*   hidx = (const int*)d_in[1];   // (B,2,E)
    const float* W    = (const float*)d_in[2]; // (C,C)
    const float* att  = (const float*)d_in[3]; // (1,H,2D)
    float* out = (float*)d_out;                // (B,N,H*D)

    // Workspace layout (floats). Zero-init regions are contiguous.
    float* ws = (float*)d_ws;
    size_t o = 0;
    float* xw      = ws + o; o += (size_t)B * N * C; // written before read
    float* alpha   = ws + o; o += (size_t)B * E * H; // written before read
    float* zbase   = ws + o;                          // --- zero block start
    float* seg_sum = ws + o; o += (size_t)B * N * H;
    float* e_feat  = ws + o; o += (size_t)B * M * C;
    float* Dnode   = ws + o; o += (size_t)B * N;
    float* Bdeg    = ws + o; o += (size_t)B * M;
    size_t zcount  = (size_t)(ws + o - zbase);        // --- zero block end
    float* seg_max = ws + o; o += (size_t)B * N * H;  // init to -inf

    const int TPB = 256;

    // Init: d_out = 0, zero block = 0, seg_max = -inf
    fill_f32<<<4096, TPB, 0, stream>>>(out, 0.0f, (size_t)B * N * C);
    fill_f32<<<4096, TPB, 0, stream>>>(zbase, 0.0f, zcount);
    fill_f32<<<512,  TPB, 0, stream>>>(seg_max, -INFINITY, (size_t)B * N * H);

    // 1) xw = x @ W  (fp32 WMMA, LDS-tiled). 128x64 macro-tile per block.
    gemm_wmma_f32<<<B * (N / 128) * (C / 64), TPB, 0, stream>>>(x, W, xw);

    // 2) degrees: B*E threads
    degrees_kernel<<<(B * E) / TPB, TPB, 0, stream>>>(hidx, Dnode, Bdeg);

    // 3) attention logits + segment max: wave per (b,e)
    alpha_logits<<<(B * E) / (TPB / 32), TPB, 0, stream>>>(xw, hidx, att, alpha, seg_max);

    // 4) exp + segment sum: thread per (b,e,h)
    exp_and_sum<<<(B * E * H) / TPB, TPB, 0, stream>>>(hidx, alpha, seg_max, seg_sum);

    // 5) normalize
    normalize_alpha<<<(B * E * H) / TPB, TPB, 0, stream>>>(hidx, alpha, seg_sum);

    // 6) msg1 scatter into e_feat
    scatter_msg1<<<(B * E) / (TPB / 32), TPB, 0, stream>>>(xw, hidx, alpha, Bdeg, e_feat);

    // 7) msg2 scatter into out
    scatter_msg2<<<(B * E) / (TPB / 32), TPB, 0, stream>>>(e_feat, hidx, alpha, Dnode, out);
}